// GCN_29850022707326
// MI455X (gfx1250) — compile-verified
//
#include <hip/hip_runtime.h>

typedef __attribute__((ext_vector_type(2))) float v2f;
typedef __attribute__((ext_vector_type(8))) float v8f;

namespace {
constexpr int N_NODES = 100000;
constexpr int N_EDGES = 1600000;
constexpr int F_IN  = 128;
constexpr int F_H   = 64;
constexpr int F_OUT = 32;
constexpr int N_G   = 128;
constexpr float BN_EPS = 1e-5f;
}

// ---- degree / normalization ------------------------------------------------
__global__ void k_fill1(float* __restrict__ p, int n) {
  int i = blockIdx.x * blockDim.x + threadIdx.x;
  if (i < n) p[i] = 1.0f;                       // self-loop contributes 1 to deg
}

__global__ void k_deg(const int* __restrict__ dst, float* __restrict__ deg) {
  int e = blockIdx.x * blockDim.x + threadIdx.x; // grid sized exactly E
  atomicAdd(&deg[dst[e]], 1.0f);
}

__global__ void k_rsqrt_inplace(float* __restrict__ p, int n) {
  int i = blockIdx.x * blockDim.x + threadIdx.x;
  if (i < n) p[i] = rsqrtf(p[i]);               // deg >= 1 always
}

// ---- dense transform: C[N,64] = A[N,K] @ W[K,64] via v_wmma_f32_16x16x4_f32
// grid.x = N/16 row tiles; block = 128 threads = 4 waves -> 4 column tiles.
__global__ void __launch_bounds__(128) k_gemm_wmma(const float* __restrict__ A,
                                                   const float* __restrict__ W,
                                                   float* __restrict__ C, int K) {
  __shared__ float sW[F_IN * F_H];              // up to 128x64 = 32 KB in LDS
  const int tid = threadIdx.x;
  for (int i = tid; i < K * F_H; i += 128) sW[i] = W[i];
  __syncthreads();

  const int wave = tid >> 5;
  const int lane = tid & 31;
  const int hi   = lane >> 4;                   // lane group (0: 0-15, 1: 16-31)
  const int m16  = lane & 15;
  const int row  = blockIdx.x * 16 + m16;       // A row for this lane
  const int koff = hi * 2;                      // lanes 16-31 hold K+2,K+3
  const int col  = wave * 16 + m16;             // B/C column

  v8f acc = {};
  const float* arow = A + (long)row * K;
  for (int k = 0; k < K; k += 4) {
    v2f a, b;
    float2 av = *(const float2*)(arow + k + koff);   // 8B aligned (k+koff even)
    a.x = av.x; a.y = av.y;
    b.x = sW[(k + koff)     * F_H + col];
    b.y = sW[(k + koff + 1) * F_H + col];
    acc = __builtin_amdgcn_wmma_f32_16x16x4_f32(false, a, false, b,
                                                (short)0, acc, false, false);
  }
  // C layout: VGPR r -> M = r + 8*hi, N = lane&15
  const int rbase = blockIdx.x * 16 + hi * 8;
#pragma unroll
  for (int r = 0; r < 8; ++r)
    C[(long)(rbase + r) * F_H + col] = acc[r];
}

// ---- acc = bias + h * dis^2   (self-loop term folded in, no atomics) -------
__global__ void __launch_bounds__(256) k_self_init(const float* __restrict__ h,
                                                   const float* __restrict__ dis,
                                                   const float* __restrict__ bias,
                                                   float* __restrict__ acc) {
  int idx = blockIdx.x * 256 + threadIdx.x;     // grid exact: N*64/256
  int c = idx & 63;
  int i = idx >> 6;
  float d = dis[i];
  acc[idx] = bias[c] + h[idx] * d * d;
}

// ---- edge scatter: one wave per edge, lanes span 64 channels ---------------
__global__ void __launch_bounds__(256) k_scatter(const float* __restrict__ h,
                                                 const float* __restrict__ dis,
                                                 const int* __restrict__ src,
                                                 const int* __restrict__ dst,
                                                 float* __restrict__ acc) {
  int wid  = (blockIdx.x * 256 + threadIdx.x) >> 5;   // edge id (grid exact)
  int lane = threadIdx.x & 31;
  int s = src[wid];
  int d = dst[wid];
  float nrm = dis[s] * dis[d];
  float m0 = h[s * 64 + lane]      * nrm;
  float m1 = h[s * 64 + lane + 32] * nrm;
  atomicAdd(&acc[d * 64 + lane],      m0);
  atomicAdd(&acc[d * 64 + lane + 32], m1);
}

// ---- batchnorm statistics: per-channel sum / sumsq -------------------------
__global__ void __launch_bounds__(256) k_bn_stats(const float* __restrict__ acc,
                                                  float* __restrict__ stats) {
  const int tid = threadIdx.x;
  const int c  = tid & 63;
  const int rg = tid >> 6;                      // 4 row groups per block
  float s = 0.f, s2 = 0.f;
  for (int i = blockIdx.x * 4 + rg; i < N_NODES; i += gridDim.x * 4) {
    float v = acc[i * 64 + c];
    s += v; s2 += v * v;
  }
  __shared__ float sh0[256], sh1[256];
  sh0[tid] = s; sh1[tid] = s2;
  __syncthreads();
  if (tid < 64) {
    s  = sh0[tid] + sh0[tid + 64] + sh0[tid + 128] + sh0[tid + 192];
    s2 = sh1[tid] + sh1[tid + 64] + sh1[tid + 128] + sh1[tid + 192];
    atomicAdd(&stats[tid], s);
    atomicAdd(&stats[64 + tid], s2);
  }
}

// ---- batchnorm apply + ReLU ------------------------------------------------
__global__ void __launch_bounds__(256) k_bn_apply(const float* __restrict__ stats,
                                                  const float* __restrict__ gamma,
                                                  const float* __restrict__ beta,
                                                  const float* __restrict__ accIn,
                                                  float* __restrict__ hOut) {
  __shared__ float ssc[64], ssh[64];
  int tid = threadIdx.x;
  if (tid < 64) {
    const float invN = 1.0f / (float)N_NODES;
    float mean = stats[tid] * invN;
    float var  = stats[64 + tid] * invN - mean * mean;   // biased variance
    float sc = gamma[tid] * rsqrtf(var + BN_EPS);
    ssc[tid] = sc;
    ssh[tid] = beta[tid] - mean * sc;
  }
  __syncthreads();
  int idx = blockIdx.x * 256 + tid;             // grid exact: N*64/256
  int c = idx & 63;
  float v = accIn[idx] * ssc[c] + ssh[c];
  hOut[idx] = v > 0.f ? v : 0.f;
}

// ---- global mean pool: wave per node, lanes span channels ------------------
__global__ void __launch_bounds__(256) k_pool(const float* __restrict__ h,
                                              const int* __restrict__ batch,
                                              float* __restrict__ pooled,
                                              float* __restrict__ cnt) {
  int node = (blockIdx.x * 256 + threadIdx.x) >> 5;     // grid exact: N waves
  int lane = threadIdx.x & 31;
  int g = batch[node];
  atomicAdd(&pooled[g * 64 + lane],      h[node * 64 + lane]);
  atomicAdd(&pooled[g * 64 + lane + 32], h[node * 64 + lane + 32]);
  if (lane == 0) atomicAdd(&cnt[g], 1.0f);
}

// ---- final linear: out[128,32] = (pooled/cnt) @ linW + linb (WMMA) ---------
// One block of 512 threads = 16 waves = all 8x2 output tiles. K = 64.
__global__ void __launch_bounds__(512) k_final_wmma(const float* __restrict__ pooled,
                                                    const float* __restrict__ cnt,
                                                    const float* __restrict__ linW,
                                                    const float* __restrict__ linb,
                                                    float* __restrict__ out) {
  __shared__ float sW[F_H * F_OUT];             // 64x32 = 8 KB
  int tid = threadIdx.x;
  for (int i = tid; i < F_H * F_OUT; i += 512) sW[i] = linW[i];
  __syncthreads();
  int wave = tid >> 5;                          // 0..15
  int lane = tid & 31;
  int rt = wave >> 1;                           // row tile 0..7
  int ct = wave & 1;                            // col tile 0..1
  int hi  = lane >> 4;
  int m16 = lane & 15;
  int row  = rt * 16 + m16;
  int koff = hi * 2;
  int col  = ct * 16 + m16;
  float invc = 1.0f / fmaxf(cnt[row], 1.0f);    // mean pool divisor
  v8f acc = {};
  for (int k = 0; k < F_H; k += 4) {
    v2f a, b;
    float2 av = *(const float2*)(pooled + row * F_H + k + koff);
    a.x = av.x * invc; a.y = av.y * invc;
    b.x = sW[(k + koff)     * F_OUT + col];
    b.y = sW[(k + koff + 1) * F_OUT + col];
    acc = __builtin_amdgcn_wmma_f32_16x16x4_f32(false, a, false, b,
                                                (short)0, acc, false, false);
  }
  float bb = linb[col];
  int rbase = rt * 16 + hi * 8;
#pragma unroll
  for (int r = 0; r < 8; ++r)
    out[(rbase + r) * F_OUT + col] = acc[r] + bb;
}

extern "C" void kernel_launch(void* const* d_in, const int* in_sizes, int n_in,
                              void* d_out, int out_size, void* d_ws, size_t ws_size,
                              hipStream_t stream) {
  (void)in_sizes; (void)n_in; (void)out_size; (void)ws_size;
  const float* x     = (const float*)d_in[0];
  const int*   ei    = (const int*)d_in[1];
  const int*   batch = (const int*)d_in[2];
  const float* W1    = (const float*)d_in[3];
  const float* b1    = (const float*)d_in[4];
  const float* W2    = (const float*)d_in[5];
  const float* b2    = (const float*)d_in[6];
  const float* gamma = (const float*)d_in[7];
  const float* beta  = (const float*)d_in[8];
  const float* linW  = (const float*)d_in[9];
  const float* linb  = (const float*)d_in[10];
  float* out = (float*)d_out;

  const int* src = ei;              // edge_index[0]
  const int* dst = ei + N_EDGES;    // edge_index[1]

  // workspace layout (floats): dis[N] | bufA[N*64] | bufB[N*64] | stats[128] |
  //                            pooled[G*64] | cnt[G]   (~52 MB total)
  float* ws     = (float*)d_ws;
  float* dis    = ws;
  float* bufA   = dis + N_NODES;
  float* bufB   = bufA + (size_t)N_NODES * F_H;
  float* stats  = bufB + (size_t)N_NODES * F_H;
  float* pooled = stats + 128;
  float* cnt    = pooled + N_G * F_H;

  // deg (with self loop) -> dis = rsqrt(deg), in place
  k_fill1<<<(N_NODES + 255) / 256, 256, 0, stream>>>(dis, N_NODES);
  k_deg<<<N_EDGES / 256, 256, 0, stream>>>(dst, dis);
  k_rsqrt_inplace<<<(N_NODES + 255) / 256, 256, 0, stream>>>(dis, N_NODES);

  // ---- layer 1 ----
  k_gemm_wmma<<<N_NODES / 16, 128, 0, stream>>>(x, W1, bufA, F_IN);
  k_self_init<<<N_NODES * F_H / 256, 256, 0, stream>>>(bufA, dis, b1, bufB);
  k_scatter<<<N_EDGES * 32 / 256, 256, 0, stream>>>(bufA, dis, src, dst, bufB);
  hipMemsetAsync(stats, 0, 128 * sizeof(float), stream);
  k_bn_stats<<<512, 256, 0, stream>>>(bufB, stats);
  k_bn_apply<<<N_NODES * F_H / 256, 256, 0, stream>>>(stats, gamma, beta, bufB, bufA);

  // ---- layer 2 ----
  k_gemm_wmma<<<N_NODES / 16, 128, 0, stream>>>(bufA, W2, bufB, F_H);
  k_self_init<<<N_NODES * F_H / 256, 256, 0, stream>>>(bufB, dis, b2, bufA);
  k_scatter<<<N_EDGES * 32 / 256, 256, 0, stream>>>(bufB, dis, src, dst, bufA);
  hipMemsetAsync(stats, 0, 128 * sizeof(float), stream);
  k_bn_stats<<<512, 256, 0, stream>>>(bufA, stats);
  k_bn_apply<<<N_NODES * F_H / 256, 256, 0, stream>>>(stats, gamma, beta, bufA, bufA);

  // ---- pool + final linear ----
  hipMemsetAsync(pooled, 0, (N_G * F_H + N_G) * sizeof(float), stream);
  k_pool<<<N_NODES * 32 / 256, 256, 0, stream>>>(bufA, batch, pooled, cnt);
  k_final_wmma<<<1, 512, 0, stream>>>(pooled, cnt, linW, linb, out);
}